// GAT_with_MLP_32358283608673
// MI455X (gfx1250) — compile-verified
//
#include <hip/hip_runtime.h>
#include <hip/hip_bf16.h>
#include <stdint.h>

typedef float v2f __attribute__((ext_vector_type(2)));
typedef float v8f __attribute__((ext_vector_type(8)));

#define HEADS 8
#define NEG_SLOPE 0.2f

static __device__ __forceinline__ unsigned orderKey(float f) {
  unsigned u = __float_as_uint(f);
  return (u & 0x80000000u) ? ~u : (u | 0x80000000u);
}
static __device__ __forceinline__ float decodeKey(unsigned u) {
  unsigned b = (u & 0x80000000u) ? (u & 0x7FFFFFFFu) : ~u;
  return __uint_as_float(b);
}
static __device__ __forceinline__ void edge_ends(const long long* __restrict__ ei,
                                                 long e, int E, int& src, int& dst) {
  if (e < E) { src = (int)ei[e]; dst = (int)ei[(long)E + e]; }
  else       { src = dst = (int)(e - E); }        // self loops appended after E
}

#define WMMA_F32(a, b, c) \
  __builtin_amdgcn_wmma_f32_16x16x4_f32(false, (a), false, (b), (short)0, (c), false, false)

// ---------------------------------------------------------------- fill ------
__global__ __launch_bounds__(256) void k_fill_u32(unsigned* __restrict__ p,
                                                  unsigned v, long n) {
  long i = (long)blockIdx.x * 256 + threadIdx.x;
  if (i < n) p[i] = v;
}

// ------------------------------------------------- fp32 WMMA GEMM -----------
// C[M,Nc] = A[M,K] @ B[K,Nc] (+ bias[col]) (optional ReLU).
// Wave computes a 16x64 tile: 4 independent accumulators share one A fragment
// per K-step (V_WMMA_F32_16X16X4_F32). K, Nc compile-time; software-pipelined
// so next step's fragments are in flight while the 4 WMMAs issue.
template <int K, int Nc>
__global__ __launch_bounds__(256) void k_gemm_wmma_f32(
    const float* __restrict__ A, const float* __restrict__ B,
    float* __restrict__ C, const float* __restrict__ bias,
    int M, int relu)
{
  const int wave = threadIdx.x >> 5;
  const int lane = threadIdx.x & 31;
  const int half = lane >> 4;          // 0: this lane carries K+0/1, 1: K+2/3
  const int l    = lane & 15;
  const int tilesN = Nc >> 6;          // 64-wide supertiles
  const int tiles  = (M >> 4) * tilesN;
  const int tile = blockIdx.x * 8 + wave;
  if (tile >= tiles) return;           // wave-uniform: EXEC stays all-ones
  const int tm = tile / tilesN;
  const int tn = tile % tilesN;

  // A: 16x4 fp32 striping — lanes 0-15 carry K+0/1, lanes 16-31 carry K+2/3.
  const float* __restrict__ ap = A + (size_t)(tm * 16 + l) * K + 2 * half;
  // B: 4x16 striping; constexpr Nc -> all 8 scalars are immediate offsets.
  const float* __restrict__ bp = B + (size_t)(2 * half) * Nc + tn * 64 + l;

  v2f a = *(const v2f*)ap;
  v2f b0, b1, b2, b3;
  b0.x = bp[0];  b0.y = bp[Nc];
  b1.x = bp[16]; b1.y = bp[16 + Nc];
  b2.x = bp[32]; b2.y = bp[32 + Nc];
  b3.x = bp[48]; b3.y = bp[48 + Nc];

  v8f acc0 = {}, acc1 = {}, acc2 = {}, acc3 = {};
#pragma unroll 4
  for (int k = 0; k < K - 4; k += 4) {
    ap += 4; bp += (size_t)4 * Nc;
    v2f an = *(const v2f*)ap;            // prefetch next step's fragments
    v2f bn0, bn1, bn2, bn3;
    bn0.x = bp[0];  bn0.y = bp[Nc];
    bn1.x = bp[16]; bn1.y = bp[16 + Nc];
    bn2.x = bp[32]; bn2.y = bp[32 + Nc];
    bn3.x = bp[48]; bn3.y = bp[48 + Nc];
    acc0 = WMMA_F32(a, b0, acc0);        // compute on previously-fetched data
    acc1 = WMMA_F32(a, b1, acc1);
    acc2 = WMMA_F32(a, b2, acc2);
    acc3 = WMMA_F32(a, b3, acc3);
    a = an; b0 = bn0; b1 = bn1; b2 = bn2; b3 = bn3;
  }
  acc0 = WMMA_F32(a, b0, acc0);
  acc1 = WMMA_F32(a, b1, acc1);
  acc2 = WMMA_F32(a, b2, acc2);
  acc3 = WMMA_F32(a, b3, acc3);

  // D: VGPR i <-> row i + 8*half, col = 16-block base + (lane&15)
  float* __restrict__ crow = C + (size_t)(tm * 16 + 8 * half) * Nc + tn * 64 + l;
  v8f accs[4] = {acc0, acc1, acc2, acc3};
#pragma unroll
  for (int j = 0; j < 4; ++j) {
    const int col = tn * 64 + j * 16 + l;
    const float bv = bias ? bias[col] : 0.0f;
#pragma unroll
    for (int i = 0; i < 8; ++i) {
      float v = accs[j][i] + bv;
      if (relu) v = fmaxf(v, 0.0f);
      crow[(size_t)i * Nc + j * 16] = v;
    }
  }
}

// -------------------------------------------- per-node attention logits -----
// h: [N, HEADS, C];  os/od: [N, HEADS]
__global__ __launch_bounds__(256) void k_node_alpha(
    const float* __restrict__ h, const float* __restrict__ a_src,
    const float* __restrict__ a_dst, float* __restrict__ os,
    float* __restrict__ od, int N, int C)
{
  int i = blockIdx.x * 256 + threadIdx.x;     // i = n*HEADS + head
  if (i >= N * HEADS) return;
  int hd = i & (HEADS - 1);
  const float* __restrict__ hp = h + (size_t)i * C;
  const float* __restrict__ as = a_src + hd * C;
  const float* __restrict__ ad = a_dst + hd * C;
  float s = 0.f, d = 0.f;
  for (int c = 0; c < C; ++c) { float v = hp[c]; s += v * as[c]; d += v * ad[c]; }
  os[i] = s; od[i] = d;
}

// ------------------------------------------------ edge pass 1: segment max --
__global__ __launch_bounds__(256) void k_edge_max(
    const long long* __restrict__ ei, const float* __restrict__ as,
    const float* __restrict__ ad, unsigned* __restrict__ mkey, int E, int N)
{
  long e = (long)blockIdx.x * 256 + threadIdx.x;
  if (e >= (long)E + N) return;
  int src, dst; edge_ends(ei, e, E, src, dst);
#pragma unroll
  for (int h = 0; h < HEADS; ++h) {
    float v = as[src * HEADS + h] + ad[dst * HEADS + h];
    v = (v >= 0.f) ? v : NEG_SLOPE * v;
    atomicMax(&mkey[dst * HEADS + h], orderKey(v));
  }
}

// --------------------------------- edge pass 2: exp + segment-sum denom -----
__global__ __launch_bounds__(256) void k_edge_exp(
    const long long* __restrict__ ei, const float* __restrict__ as,
    const float* __restrict__ ad, const unsigned* __restrict__ mkey,
    float* __restrict__ denom, float* __restrict__ ex, int E, int N)
{
  long e = (long)blockIdx.x * 256 + threadIdx.x;
  if (e >= (long)E + N) return;
  int src, dst; edge_ends(ei, e, E, src, dst);
#pragma unroll
  for (int h = 0; h < HEADS; ++h) {
    float v = as[src * HEADS + h] + ad[dst * HEADS + h];
    v = (v >= 0.f) ? v : NEG_SLOPE * v;
    float m = decodeKey(mkey[dst * HEADS + h]);
    float x = __expf(v - m);
    ex[e * HEADS + h] = x;
    atomicAdd(&denom[dst * HEADS + h], x);
  }
}

// --------------------------- edge pass 3: gather * alpha, scatter-add -------
// One wave per edge; lanes stride the HC channels (coalesced source row read).
__global__ __launch_bounds__(256) void k_edge_aggregate(
    const long long* __restrict__ ei, const float* __restrict__ h,
    const float* __restrict__ ex, const float* __restrict__ denom,
    float* __restrict__ agg, int E, int N, int HC, int logC)
{
  int wave = threadIdx.x >> 5, lane = threadIdx.x & 31;
  long e = (long)blockIdx.x * 8 + wave;
  if (e >= (long)E + N) return;
  int src, dst; edge_ends(ei, e, E, src, dst);
  const float* __restrict__ hs  = h   + (size_t)src * HC;
  float* __restrict__       ag  = agg + (size_t)dst * HC;
  const float* __restrict__ exe = ex  + e * HEADS;
  for (int c = lane; c < HC; c += 32) {
    int hd = c >> logC;
    float alpha = exe[hd] / denom[dst * HEADS + hd];
    atomicAdd(&ag[c], hs[c] * alpha);
  }
}

// ------------------------------------------------- relu(agg + bias) ---------
__global__ __launch_bounds__(256) void k_relu_bias(
    const float* __restrict__ agg, const float* __restrict__ bias,
    float* __restrict__ out, long n, int HC)
{
  long i = (long)blockIdx.x * 256 + threadIdx.x;
  if (i >= n) return;
  out[i] = fmaxf(agg[i] + bias[(int)(i % HC)], 0.f);
}

// ------------------------------------- mean over heads + b2  ([N,8,64]) -----
__global__ __launch_bounds__(256) void k_head_mean(
    const float* __restrict__ agg, const float* __restrict__ b2,
    float* __restrict__ xm, int N)
{
  int i = blockIdx.x * 256 + threadIdx.x;     // i = n*64 + c
  if (i >= N * 64) return;
  int n = i >> 6, c = i & 63;
  const float* __restrict__ p = agg + (size_t)n * 512 + c;
  float s = 0.f;
#pragma unroll
  for (int h = 0; h < HEADS; ++h) s += p[h * 64];
  xm[i] = s * 0.125f + b2[c];
}

// ------------------------------------------ final [64,2] projection ---------
__global__ __launch_bounds__(256) void k_mlp_out(
    const float* __restrict__ hm, const float* __restrict__ Wm2,
    const float* __restrict__ bm2, float* __restrict__ out, int N)
{
  int n = blockIdx.x * 256 + threadIdx.x;
  if (n >= N) return;
  const float* __restrict__ r = hm + (size_t)n * 64;
  float o0 = bm2[0], o1 = bm2[1];
#pragma unroll 8
  for (int k = 0; k < 64; ++k) { float v = r[k]; o0 += v * Wm2[k * 2]; o1 += v * Wm2[k * 2 + 1]; }
  out[n * 2]     = o0;
  out[n * 2 + 1] = o1;
}

// ============================================================ launcher ======
#define CDIV(a, b) (((a) + (b) - 1) / (b))

extern "C" void kernel_launch(void* const* d_in, const int* in_sizes, int n_in,
                              void* d_out, int out_size, void* d_ws, size_t ws_size,
                              hipStream_t stream) {
  const float*     x   = (const float*)d_in[0];
  const long long* ei  = (const long long*)d_in[1];   // int64 [2,E]
  const float*     W1  = (const float*)d_in[2];
  const float*     a1s = (const float*)d_in[3];
  const float*     a1d = (const float*)d_in[4];
  const float*     b1  = (const float*)d_in[5];
  const float*     W2  = (const float*)d_in[6];
  const float*     a2s = (const float*)d_in[7];
  const float*     a2d = (const float*)d_in[8];
  const float*     b2  = (const float*)d_in[9];
  const float*     Wm1 = (const float*)d_in[10];
  const float*     bm1 = (const float*)d_in[11];
  const float*     Wm2 = (const float*)d_in[12];
  const float*     bm2 = (const float*)d_in[13];
  float* out = (float*)d_out;

  const int N  = in_sizes[0] / 128;     // 50000
  const int E  = in_sizes[1] / 2;       // 400000
  const long ET = (long)E + N;          // + self loops
  const size_t NN = (size_t)N;

  // ---- workspace carve-out (fp32 elements) ----
  float* ws   = (float*)d_ws;
  float* h1    = ws;                        // [N,256] gemm1 out, later relu(agg1+b1)
  float* agg1  = h1   + NN * 256;           // [N,256] layer-1 aggregate; reused as MLP hidden
  float* h2    = agg1 + NN * 256;           // [N,512] gemm2 out
  float* agg2  = h2   + NN * 512;           // [N,512] layer-2 aggregate
  float* as_   = agg2 + NN * 512;           // [N,8]
  float* ad_   = as_  + NN * 8;             // [N,8]
  unsigned* mkey = (unsigned*)(ad_ + NN * 8);  // [N,8] ordered-max keys
  float* denom = (float*)(mkey + NN * 8);   // [N,8]
  float* ex    = denom + NN * 8;            // [E+N,8]
  float* xm    = ex + (size_t)ET * 8;       // [N,64] head mean
  float* hm    = agg1;                      // [N,64] MLP hidden (agg1 is free by then)

  const int thr = 256;

  // =================== GAT layer 1 (heads=8, C=32, concat) ===================
  {
    int tiles = (N / 16) * (256 / 64);
    k_gemm_wmma_f32<128, 256><<<CDIV(tiles, 8), thr, 0, stream>>>(x, W1, h1, nullptr, N, 0);
    k_node_alpha<<<CDIV(N * HEADS, thr), thr, 0, stream>>>(h1, a1s, a1d, as_, ad_, N, 32);
    k_fill_u32<<<CDIV((long)N * 8, (long)thr), thr, 0, stream>>>(mkey, 0u, (long)N * 8);
    k_fill_u32<<<CDIV((long)N * 8, (long)thr), thr, 0, stream>>>((unsigned*)denom, 0u, (long)N * 8);
    k_fill_u32<<<CDIV((long)N * 256, (long)thr), thr, 0, stream>>>((unsigned*)agg1, 0u, (long)N * 256);
    k_edge_max<<<CDIV(ET, (long)thr), thr, 0, stream>>>(ei, as_, ad_, mkey, E, N);
    k_edge_exp<<<CDIV(ET, (long)thr), thr, 0, stream>>>(ei, as_, ad_, mkey, denom, ex, E, N);
    k_edge_aggregate<<<CDIV(ET, 8L), thr, 0, stream>>>(ei, h1, ex, denom, agg1, E, N, 256, 5);
    k_relu_bias<<<CDIV((long)N * 256, (long)thr), thr, 0, stream>>>(agg1, b1, h1, (long)N * 256, 256);
  }

  // =================== GAT layer 2 (heads=8, C=64, mean) =====================
  {
    int tiles = (N / 16) * (512 / 64);
    k_gemm_wmma_f32<256, 512><<<CDIV(tiles, 8), thr, 0, stream>>>(h1, W2, h2, nullptr, N, 0);
    k_node_alpha<<<CDIV(N * HEADS, thr), thr, 0, stream>>>(h2, a2s, a2d, as_, ad_, N, 64);
    k_fill_u32<<<CDIV((long)N * 8, (long)thr), thr, 0, stream>>>(mkey, 0u, (long)N * 8);
    k_fill_u32<<<CDIV((long)N * 8, (long)thr), thr, 0, stream>>>((unsigned*)denom, 0u, (long)N * 8);
    k_fill_u32<<<CDIV((long)N * 512, (long)thr), thr, 0, stream>>>((unsigned*)agg2, 0u, (long)N * 512);
    k_edge_max<<<CDIV(ET, (long)thr), thr, 0, stream>>>(ei, as_, ad_, mkey, E, N);
    k_edge_exp<<<CDIV(ET, (long)thr), thr, 0, stream>>>(ei, as_, ad_, mkey, denom, ex, E, N);
    k_edge_aggregate<<<CDIV(ET, 8L), thr, 0, stream>>>(ei, h2, ex, denom, agg2, E, N, 512, 6);
    k_head_mean<<<CDIV(N * 64, thr), thr, 0, stream>>>(agg2, b2, xm, N);
  }

  // =================== MLP ===================================================
  {
    int tiles = (N / 16) * (64 / 64);
    k_gemm_wmma_f32<64, 64><<<CDIV(tiles, 8), thr, 0, stream>>>(xm, Wm1, hm, bm1, N, 1);
    k_mlp_out<<<CDIV(N, thr), thr, 0, stream>>>(hm, Wm2, bm2, out, N);
  }
}